// TransformerAttentionModule_10273561772506
// MI455X (gfx1250) — compile-verified
//
#include <hip/hip_runtime.h>
#include <math.h>

#define DIM 128
#define NHEADS 8
#define HDIM 16

typedef __attribute__((ext_vector_type(2))) float v2f;
typedef __attribute__((ext_vector_type(8))) float v8f;

// ---- monotone float<->uint mapping so float-max can use atomicMax(uint) ----
__device__ __forceinline__ unsigned mapf(float f) {
    unsigned u = __float_as_uint(f);
    return (u & 0x80000000u) ? ~u : (u | 0x80000000u);
}
__device__ __forceinline__ float unmapf(unsigned m) {
    return (m & 0x80000000u) ? __uint_as_float(m ^ 0x80000000u)
                             : __uint_as_float(~m);
}
#define NEG_INF_MAPPED 0x007FFFFFu   // mapf(-inf)

// --------------------------------------------------------------------------
// init: zero aggregation + denom, seed per-(node,head) running max
// --------------------------------------------------------------------------
__global__ void gat_init_kernel(float* __restrict__ agg, float* __restrict__ denom,
                                unsigned* __restrict__ nmax, int n) {
    int i = blockIdx.x * blockDim.x + threadIdx.x;
    if (i < n * DIM) agg[i] = 0.0f;
    if (i < n * NHEADS) { denom[i] = 0.0f; nmax[i] = NEG_INF_MAPPED; }
}

// --------------------------------------------------------------------------
// proj: q = 0.25*(h@Wq^T + bq), k = h@Wk^T + bk, v = h@Wv^T + bv
// One block = 16-node tile staged in LDS; wave w owns output columns
// [16w,16w+16) and accumulates all three matrices with f32 WMMA.
// A frag (16x4, M=lane%16): K = 4*ks + 2*(lane/16) + {0,1}
// B frag (4x16, N=lane%16): same K indexing; B[k][n] = W[n][k] (W row-major)
// C/D (16x16): VGPR j -> M = j + 8*(lane/16), N = lane%16
// --------------------------------------------------------------------------
__global__ __launch_bounds__(256)
void gat_proj_kernel(const float* __restrict__ h,
                     const float* __restrict__ Wq, const float* __restrict__ bq,
                     const float* __restrict__ Wk, const float* __restrict__ bk,
                     const float* __restrict__ Wv, const float* __restrict__ bv,
                     float* __restrict__ q, float* __restrict__ k, float* __restrict__ v,
                     int n) {
    __shared__ float sh[16 * DIM];
    const int tid  = threadIdx.x;
    const int tile = blockIdx.x * 16;

    // cooperative stage of 16x128 h tile (rows past n -> 0)
    for (int j = 0; j < 8; ++j) {
        int fi = j * 256 + tid;
        int r = fi >> 7, c = fi & 127;
        int gr = tile + r;
        sh[fi] = (gr < n) ? h[gr * DIM + c] : 0.0f;
    }
    __syncthreads();

    const int wave = tid >> 5;
    const int lane = tid & 31;
    const int half = lane >> 4;
    const int ln16 = lane & 15;
    const int dcol = wave * 16 + ln16;        // output column (N)

    v8f aq = {}; v8f ak = {}; v8f av = {};
    const int arow = ln16 * DIM;
    const int brow = dcol * DIM;
    for (int ks = 0; ks < 32; ++ks) {
        const int koff = ks * 4 + 2 * half;
        v2f a   = *(const v2f*)&sh[arow + koff];
        v2f bqf = *(const v2f*)&Wq[brow + koff];
        v2f bkf = *(const v2f*)&Wk[brow + koff];
        v2f bvf = *(const v2f*)&Wv[brow + koff];
        aq = __builtin_amdgcn_wmma_f32_16x16x4_f32(false, a, false, bqf, (short)0, aq, false, false);
        ak = __builtin_amdgcn_wmma_f32_16x16x4_f32(false, a, false, bkf, (short)0, ak, false, false);
        av = __builtin_amdgcn_wmma_f32_16x16x4_f32(false, a, false, bvf, (short)0, av, false, false);
    }

    const float cbq = bq[dcol], cbk = bk[dcol], cbv = bv[dcol];
    for (int j = 0; j < 8; ++j) {
        int row = tile + j + 8 * half;
        if (row < n) {
            q[row * DIM + dcol] = (aq[j] + cbq) * 0.25f;  // fold 1/sqrt(HDIM)
            k[row * DIM + dcol] =  ak[j] + cbk;
            v[row * DIM + dcol] =  av[j] + cbv;
        }
    }
}

// --------------------------------------------------------------------------
// scores: one thread per (edge, head): s = dot16(q[src], k[dst]) (scale in q)
// plus running segment-max via mapped-uint atomicMax
// --------------------------------------------------------------------------
__global__ void gat_scores_kernel(const float* __restrict__ q, const float* __restrict__ k,
                                  const int* __restrict__ src, const int* __restrict__ dst,
                                  float* __restrict__ s, unsigned* __restrict__ nmax, int nE) {
    int t = blockIdx.x * blockDim.x + threadIdx.x;
    if (t >= nE * NHEADS) return;
    int e  = t >> 3;
    int hd = t & 7;
    int sn = src[e], dn = dst[e];
    const float4* qp = (const float4*)&q[sn * DIM + hd * HDIM];
    const float4* kp = (const float4*)&k[dn * DIM + hd * HDIM];
    float acc = 0.0f;
    #pragma unroll
    for (int i = 0; i < 4; ++i) {
        float4 a = qp[i], b = kp[i];
        acc += a.x * b.x + a.y * b.y + a.z * b.z + a.w * b.w;
    }
    s[t] = acc;
    atomicMax(&nmax[dn * NHEADS + hd], mapf(acc));
}

// --------------------------------------------------------------------------
// expsum: e = exp(s - m[dst]); overwrite s; atomic segment-sum into denom
// --------------------------------------------------------------------------
__global__ void gat_expsum_kernel(const int* __restrict__ dst, float* __restrict__ s,
                                  const unsigned* __restrict__ nmax,
                                  float* __restrict__ denom, int nE) {
    int t = blockIdx.x * blockDim.x + threadIdx.x;
    if (t >= nE * NHEADS) return;
    int e  = t >> 3;
    int hd = t & 7;
    int dn = dst[e];
    float m  = unmapf(nmax[dn * NHEADS + hd]);
    float ev = __expf(s[t] - m);
    s[t] = ev;
    atomicAdd(&denom[dn * NHEADS + hd], ev);
}

// denom -> 1/denom (isolated nodes give inf, never read)
__global__ void gat_rcp_kernel(float* __restrict__ denom, int total) {
    int i = blockIdx.x * blockDim.x + threadIdx.x;
    if (i < total) denom[i] = 1.0f / denom[i];
}

// --------------------------------------------------------------------------
// agg: one thread per (edge, dim): agg[dst][d] += (e/denom) * v[src][d]
// --------------------------------------------------------------------------
__global__ void gat_agg_kernel(const int* __restrict__ src, const int* __restrict__ dst,
                               const float* __restrict__ s, const float* __restrict__ rden,
                               const float* __restrict__ v, float* __restrict__ agg,
                               long long total) {
    long long t = (long long)blockIdx.x * blockDim.x + threadIdx.x;
    if (t >= total) return;
    int e  = (int)(t >> 7);
    int d  = (int)(t & 127);
    int hd = d >> 4;
    int sn = src[e], dn = dst[e];
    float p = s[e * NHEADS + hd] * rden[dn * NHEADS + hd];
    atomicAdd(&agg[dn * DIM + d], p * v[sn * DIM + d]);
}

// --------------------------------------------------------------------------
// out: out = agg @ Wo^T + bo  (same WMMA tiling as proj, single matrix)
// --------------------------------------------------------------------------
__global__ __launch_bounds__(256)
void gat_out_kernel(const float* __restrict__ agg, const float* __restrict__ Wo,
                    const float* __restrict__ bo, float* __restrict__ out, int n) {
    __shared__ float sh[16 * DIM];
    const int tid  = threadIdx.x;
    const int tile = blockIdx.x * 16;

    for (int j = 0; j < 8; ++j) {
        int fi = j * 256 + tid;
        int r = fi >> 7, c = fi & 127;
        int gr = tile + r;
        sh[fi] = (gr < n) ? agg[gr * DIM + c] : 0.0f;
    }
    __syncthreads();

    const int wave = tid >> 5;
    const int lane = tid & 31;
    const int half = lane >> 4;
    const int ln16 = lane & 15;
    const int dcol = wave * 16 + ln16;

    v8f acc = {};
    const int arow = ln16 * DIM;
    const int brow = dcol * DIM;
    for (int ks = 0; ks < 32; ++ks) {
        const int koff = ks * 4 + 2 * half;
        v2f a  = *(const v2f*)&sh[arow + koff];
        v2f bf = *(const v2f*)&Wo[brow + koff];
        acc = __builtin_amdgcn_wmma_f32_16x16x4_f32(false, a, false, bf, (short)0, acc, false, false);
    }
    const float cb = bo[dcol];
    for (int j = 0; j < 8; ++j) {
        int row = tile + j + 8 * half;
        if (row < n) out[row * DIM + dcol] = acc[j] + cb;
    }
}

// --------------------------------------------------------------------------
extern "C" void kernel_launch(void* const* d_in, const int* in_sizes, int n_in,
                              void* d_out, int out_size, void* d_ws, size_t ws_size,
                              hipStream_t stream) {
    const float* h   = (const float*)d_in[0];
    const int*   src = (const int*)  d_in[1];
    const int*   dst = (const int*)  d_in[2];
    const float* Wq  = (const float*)d_in[3];
    const float* bq  = (const float*)d_in[4];
    const float* Wk  = (const float*)d_in[5];
    const float* bk  = (const float*)d_in[6];
    const float* Wv  = (const float*)d_in[7];
    const float* bv  = (const float*)d_in[8];
    const float* Wo  = (const float*)d_in[9];
    const float* bo  = (const float*)d_in[10];
    float* out = (float*)d_out;

    const int n  = in_sizes[0] / DIM;   // 50000
    const int nE = in_sizes[1];         // 800000

    // workspace layout (floats): q | k | v | scores/e | agg | denom | nmax
    const size_t nd = (size_t)n * DIM;
    const size_t e8 = (size_t)nE * NHEADS;
    float*    q_buf  = (float*)d_ws;
    float*    k_buf  = q_buf + nd;
    float*    v_buf  = k_buf + nd;
    float*    s_buf  = v_buf + nd;
    float*    agg    = s_buf + e8;
    float*    denom  = agg   + nd;
    unsigned* nmax   = (unsigned*)(denom + (size_t)n * NHEADS);

    const int tiles = (n + 15) / 16;

    // 1) init reductions
    {
        int total = n * DIM;
        gat_init_kernel<<<(total + 255) / 256, 256, 0, stream>>>(agg, denom, nmax, n);
    }
    // 2) projections (WMMA f32)
    gat_proj_kernel<<<tiles, 256, 0, stream>>>(h, Wq, bq, Wk, bk, Wv, bv,
                                               q_buf, k_buf, v_buf, n);
    // 3) edge scores + segment max
    {
        int total = nE * NHEADS;
        gat_scores_kernel<<<(total + 255) / 256, 256, 0, stream>>>(q_buf, k_buf, src, dst,
                                                                   s_buf, nmax, nE);
        gat_expsum_kernel<<<(total + 255) / 256, 256, 0, stream>>>(dst, s_buf, nmax, denom, nE);
    }
    // 4) reciprocal of denominators
    {
        int total = n * NHEADS;
        gat_rcp_kernel<<<(total + 255) / 256, 256, 0, stream>>>(denom, total);
    }
    // 5) weighted aggregation (scatter atomics)
    {
        long long total = (long long)nE * DIM;
        int blocks = (int)((total + 255) / 256);
        gat_agg_kernel<<<blocks, 256, 0, stream>>>(src, dst, s_buf, denom, v_buf, agg, total);
    }
    // 6) output projection (WMMA f32)
    gat_out_kernel<<<tiles, 256, 0, stream>>>(agg, Wo, bo, out, n);
}